// TopKLoss_48326972015079
// MI455X (gfx1250) — compile-verified
//
#include <hip/hip_runtime.h>
#include <stdint.h>

// ---------------------------------------------------------------------------
// TopK-mean of BCEWithLogits loss, exact 2-level radix select (16+16 bits)
// on the (nonnegative -> order-preserving) float bit pattern.
// Pass A: 65536-bin LDS histogram of top 16 bits       (256KB LDS, CDNA5-only)
// Pass B: 65536-bin LDS histogram of low 16 bits of the candidate bin
// Pass C: sum of losses strictly above threshold, TDM-staged through LDS
// Combine: mean = (sum_gt + (k - cnt_gt) * T) / k   (exact tie handling)
// ---------------------------------------------------------------------------

#define HIST_BINS   65536
#define HIST_TPB    1024
#define SUM_TILE    4096        // floats per TDM tile (16KB)
#define SUM_TPB     256
#define SUM_GRID    512
#define HIST_GRID   512

// loss = logaddexp(0,x) - x*t = max(x,0) + log1p(exp(-|x|)) - x*t  (>= 0 in FP)
__device__ __forceinline__ float bce_loss(float x, float t) {
    return fmaxf(x, 0.0f) + log1pf(__expf(-fabsf(x))) - x * t;
}

// Generic LDS pointers are {SHARED_BASE aperture, byte offset}; low 32 bits are
// the wave-relative LDS byte address the TDM descriptor wants.
__device__ __forceinline__ uint32_t lds_byte_off(const void* p) {
    return (uint32_t)(uintptr_t)p;
}

typedef unsigned int u32x4 __attribute__((ext_vector_type(4)));
typedef int          i32x8 __attribute__((ext_vector_type(8)));
typedef int          i32x4 __attribute__((ext_vector_type(4)));

// Issue a 1-D TENSOR_LOAD_TO_LDS of `nelem` f32 from gptr into LDS offset.
// D# per CDNA5 ISA ch.8: group0 = {count=1 | lds_addr | global_addr | type=2},
// group1 = {data_size=4B, tensor_dim0=tile_dim0=nelem, stride=nelem}.
// clang-23 toolchain exposes the 6-arg builtin:
//   (u32x4 g0, i32x8 g1, i32x4 g2, i32x4 g3, i32x8 extra, i32 cpol)
__device__ __forceinline__ void tdm_load_f32_1d(uint32_t lds_off,
                                                const float* gptr,
                                                uint32_t nelem) {
    uint64_t ga = (uint64_t)(uintptr_t)gptr;
    u32x4 g0;
    g0.x = 1u;                                           // count=1, user D#
    g0.y = lds_off;                                      // lds_addr (bytes)
    g0.z = (uint32_t)ga;                                 // global_addr[31:0]
    g0.w = (uint32_t)((ga >> 32) & 0x01FFFFFFu)          // global_addr[56:32]
         | (2u << 30);                                   // type = 2 ("image")
    i32x8 g1;
    g1[0] = (int)(2u << 16);                             // wg_mask=0, data_size=4B
    g1[1] = (int)((nelem & 0xFFFFu) << 16);              // tensor_dim0[15:0]  @63:48
    g1[2] = (int)((nelem >> 16) & 0xFFFFu);              // tensor_dim0[31:16] @79:64
    g1[3] = (int)((nelem & 0xFFFFu) << 16);              // tile_dim0 @127:112
    g1[4] = 0;                                           // tile_dim1/2 unused
    g1[5] = (int)nelem;                                  // tensor_dim0_stride lo
    g1[6] = 0;
    g1[7] = 0;
    i32x4 z4 = {0, 0, 0, 0};                             // groups 2/3 unused (<=2D)
    i32x8 z8 = {0, 0, 0, 0, 0, 0, 0, 0};
    __builtin_amdgcn_tensor_load_to_lds(g0, g1, z4, z4, z8, 0);
}

// --- workspace zeroing (u32 words), grid-strided ---------------------------
__global__ void zero_kernel(uint32_t* p, size_t n) {
    size_t i = (size_t)blockIdx.x * blockDim.x + threadIdx.x;
    size_t stride = (size_t)gridDim.x * blockDim.x;
    for (; i < n; i += stride) p[i] = 0u;
}

// --- histogram pass: level 1 -> top 16 bits; level 2 -> low 16 bits of bin --
__global__ __launch_bounds__(HIST_TPB)
void hist_kernel(const float4* __restrict__ x, const float4* __restrict__ t,
                 uint32_t* __restrict__ ghist, size_t n4,
                 const uint32_t* __restrict__ ctrl, int level) {
    __shared__ uint32_t h[HIST_BINS];                    // 256KB: needs CDNA5 WGP LDS
    for (int i = threadIdx.x; i < HIST_BINS; i += HIST_TPB) h[i] = 0u;
    uint32_t topsel = ctrl[0];                           // only meaningful at level 2
    __syncthreads();

    size_t i = (size_t)blockIdx.x * HIST_TPB + threadIdx.x;
    size_t stride = (size_t)gridDim.x * HIST_TPB;
    for (; i < n4; i += stride) {
        float4 xv = x[i];
        float4 tv = t[i];
        float l0 = bce_loss(xv.x, tv.x), l1 = bce_loss(xv.y, tv.y);
        float l2 = bce_loss(xv.z, tv.z), l3 = bce_loss(xv.w, tv.w);
        uint32_t b0 = __float_as_uint(l0), b1 = __float_as_uint(l1);
        uint32_t b2 = __float_as_uint(l2), b3 = __float_as_uint(l3);
        if (level == 1) {
            atomicAdd(&h[b0 >> 16], 1u);
            atomicAdd(&h[b1 >> 16], 1u);
            atomicAdd(&h[b2 >> 16], 1u);
            atomicAdd(&h[b3 >> 16], 1u);
        } else {
            if ((b0 >> 16) == topsel) atomicAdd(&h[b0 & 0xFFFFu], 1u);
            if ((b1 >> 16) == topsel) atomicAdd(&h[b1 & 0xFFFFu], 1u);
            if ((b2 >> 16) == topsel) atomicAdd(&h[b2 & 0xFFFFu], 1u);
            if ((b3 >> 16) == topsel) atomicAdd(&h[b3 & 0xFFFFu], 1u);
        }
    }
    __syncthreads();
    for (int i2 = threadIdx.x; i2 < HIST_BINS; i2 += HIST_TPB) {
        uint32_t v = h[i2];
        if (v) atomicAdd(&ghist[i2], v);
    }
}

// --- descending-rank selection over a 65536-bin histogram -------------------
// level 1: ctrl[0]=bin, ctrl[1]=count strictly above bin
// level 2: ctrl[2]=threshold bits, ctrl[3]=count strictly greater than T
__global__ __launch_bounds__(1024)
void select_kernel(const uint32_t* __restrict__ hist, uint32_t* ctrl,
                   uint32_t k, int level) {
    __shared__ uint32_t csum[1024];
    __shared__ uint32_t csuf[1024];
    uint32_t rank = (level == 1) ? k : (k - ctrl[1]);
    int t = threadIdx.x;
    uint32_t s = 0;
    for (int b = t * 64; b < t * 64 + 64; ++b) s += hist[b];
    csum[t] = s;
    __syncthreads();
    if (t == 0) {                                        // suffix over 1024 chunks
        uint32_t run = 0;
        for (int c = 1023; c >= 0; --c) { csuf[c] = run; run += csum[c]; }
    }
    __syncthreads();
    uint32_t run = csuf[t];
    for (int b = t * 64 + 63; b >= t * 64; --b) {
        uint32_t hv = hist[b];
        if (run < rank && run + hv >= rank) {            // exactly one thread/bin hits
            if (level == 1) { ctrl[0] = (uint32_t)b; ctrl[1] = run; }
            else {
                ctrl[2] = (ctrl[0] << 16) | (uint32_t)b;
                ctrl[3] = ctrl[1] + run;
            }
        }
        run += hv;
    }
}

// --- sum of losses strictly > threshold; TDM double-buffered through LDS ----
__global__ __launch_bounds__(SUM_TPB)
void sum_kernel(const float* __restrict__ x, const float* __restrict__ t,
                const uint32_t* __restrict__ ctrl, float* __restrict__ partials,
                int nTiles) {
    __shared__ float sx[2][SUM_TILE];
    __shared__ float st[2][SUM_TILE];
    __shared__ float red[SUM_TPB];
    const uint32_t T = ctrl[2];
    const bool w0 = (threadIdx.x < 32);                  // one TDM issue per block
    float acc = 0.0f;

    int tile = blockIdx.x;
    if (w0 && tile < nTiles) {                           // prologue: stage tile 0
        size_t base = (size_t)tile * SUM_TILE;
        tdm_load_f32_1d(lds_byte_off(&sx[0][0]), x + base, SUM_TILE);
        tdm_load_f32_1d(lds_byte_off(&st[0][0]), t + base, SUM_TILE);
    }
    int buf = 0;
    for (; tile < nTiles; tile += gridDim.x) {
        int nxt = tile + gridDim.x;
        if (w0) {
            if (nxt < nTiles) {                          // stage next tile into buf^1
                size_t base = (size_t)nxt * SUM_TILE;
                tdm_load_f32_1d(lds_byte_off(&sx[buf ^ 1][0]), x + base, SUM_TILE);
                tdm_load_f32_1d(lds_byte_off(&st[buf ^ 1][0]), t + base, SUM_TILE);
                __builtin_amdgcn_s_wait_tensorcnt(2);    // current tile's pair done
            } else {
                __builtin_amdgcn_s_wait_tensorcnt(0);    // drain
            }
        }
        __syncthreads();                                 // publish LDS tile to all waves
#pragma unroll
        for (int j = 0; j < SUM_TILE / SUM_TPB; ++j) {
            int i = threadIdx.x + j * SUM_TPB;
            float l = bce_loss(sx[buf][i], st[buf][i]);
            if (__float_as_uint(l) > T) acc += l;
        }
        __syncthreads();                                 // done reading before reuse
        buf ^= 1;
    }

    red[threadIdx.x] = acc;
    __syncthreads();
    for (int sft = SUM_TPB / 2; sft > 0; sft >>= 1) {    // deterministic tree reduce
        if (threadIdx.x < sft) red[threadIdx.x] += red[threadIdx.x + sft];
        __syncthreads();
    }
    if (threadIdx.x == 0) partials[blockIdx.x] = red[0];
}

// --- deterministic final combine -------------------------------------------
__global__ void final_kernel(const uint32_t* ctrl, const float* partials,
                             int nPart, float* out, uint32_t k) {
    if (threadIdx.x == 0 && blockIdx.x == 0) {
        double s = 0.0;
        for (int i = 0; i < nPart; ++i) s += (double)partials[i];
        uint32_t T = ctrl[2];
        uint32_t cg = ctrl[3];                           // count strictly > T
        double res = (s + (double)(k - cg) * (double)__uint_as_float(T))
                   / (double)k;
        out[0] = (float)res;
    }
}

extern "C" void kernel_launch(void* const* d_in, const int* in_sizes, int n_in,
                              void* d_out, int out_size, void* d_ws, size_t ws_size,
                              hipStream_t stream) {
    const float* x = (const float*)d_in[0];
    const float* t = (const float*)d_in[1];
    float* out = (float*)d_out;

    size_t N = (size_t)in_sizes[0];                      // 33,554,432
    uint32_t k = (uint32_t)(N / 10);                     // int(N * 0.1) = 3,355,443

    // workspace layout (u32 words): [hist:65536][ctrl:8][partials:512]
    uint32_t* hist = (uint32_t*)d_ws;
    uint32_t* ctrl = hist + HIST_BINS;
    float* partials = (float*)(ctrl + 8);

    zero_kernel<<<256, 256, 0, stream>>>(hist, (size_t)(HIST_BINS + 8 + SUM_GRID));
    hist_kernel<<<HIST_GRID, HIST_TPB, 0, stream>>>(
        (const float4*)x, (const float4*)t, hist, N / 4, ctrl, 1);
    select_kernel<<<1, 1024, 0, stream>>>(hist, ctrl, k, 1);
    zero_kernel<<<256, 256, 0, stream>>>(hist, (size_t)HIST_BINS);
    hist_kernel<<<HIST_GRID, HIST_TPB, 0, stream>>>(
        (const float4*)x, (const float4*)t, hist, N / 4, ctrl, 2);
    select_kernel<<<1, 1024, 0, stream>>>(hist, ctrl, k, 2);
    sum_kernel<<<SUM_GRID, SUM_TPB, 0, stream>>>(x, t, ctrl, partials,
                                                 (int)(N / SUM_TILE));
    final_kernel<<<1, 1, 0, stream>>>(ctrl, partials, SUM_GRID, out, k);
}